// GCNBlock_48086453846714
// MI455X (gfx1250) — compile-verified
//
#include <hip/hip_runtime.h>
#include <hip/hip_bf16.h>

typedef __attribute__((ext_vector_type(2))) float v2f;
typedef __attribute__((ext_vector_type(8))) float v8f;

#define IN_CH  128
#define OUT_CH 128

// ---------------------------------------------------------------------------
// 1) degree init: self-loop contributes 1 to every node's degree
__global__ void gcn_deg_init(unsigned* __restrict__ deg, int N) {
    int i = blockIdx.x * blockDim.x + threadIdx.x;
    if (i < N) deg[i] = 1u;
}

// 2) degree accumulate over edge destinations
__global__ void gcn_deg_accum(const int* __restrict__ dst, unsigned* __restrict__ deg, int E) {
    int e = blockIdx.x * blockDim.x + threadIdx.x;
    if (e < E) atomicAdd(&deg[dst[e]], 1u);
}

// 3) dinv = rsqrt(deg)  (deg >= 1 always due to self loops)
__global__ void gcn_dinv(const unsigned* __restrict__ deg, float* __restrict__ dinv, int N) {
    int i = blockIdx.x * blockDim.x + threadIdx.x;
    if (i < N) {
        unsigned d = deg[i];
        dinv[i] = d ? rsqrtf((float)d) : 0.0f;
    }
}

// ---------------------------------------------------------------------------
// 4) xw = x @ W  via V_WMMA_F32_16X16X4_F32.
// One wave -> one 16x16 output tile; 8 waves/block -> 16 rows x 128 cols.
// A frag (16x4 f32): lanes 0-15 hold M=0-15 K={k0,k0+1}; lanes 16-31 K={k0+2,k0+3}.
// B frag (4x16 f32): lanes 0-15 hold N=0-15 K=k0 rows {0,1}; lanes 16-31 rows {2,3}.
// C/D (16x16 f32): VGPR r: lanes 0-15 -> M=r, lanes 16-31 -> M=r+8; N = lane%16.
__global__ void __launch_bounds__(256)
gcn_xw_wmma(const float* __restrict__ x, const float* __restrict__ W,
            float* __restrict__ xw, int N) {
    const int wave = threadIdx.x >> 5;     // 0..7 -> column tile
    const int lane = threadIdx.x & 31;
    const int row0 = blockIdx.x * 16;
    const int col0 = wave * 16;
    if (row0 >= N) return;                  // uniform per block

    const int mn    = lane & 15;            // M for A, N for B, N for C/D
    const int khalf = (lane >> 4) << 1;     // 0 or 2

    const float* __restrict__ xrow = x + (size_t)(row0 + mn) * IN_CH;

    v8f acc = {};
#pragma unroll
    for (int k0 = 0; k0 < IN_CH; k0 += 4) {
        v2f a, b;
        a.x = xrow[k0 + khalf];
        a.y = xrow[k0 + khalf + 1];
        b.x = W[(size_t)(k0 + khalf) * OUT_CH + col0 + mn];
        b.y = W[(size_t)(k0 + khalf + 1) * OUT_CH + col0 + mn];
        acc = __builtin_amdgcn_wmma_f32_16x16x4_f32(
            /*neg_a=*/false, a, /*neg_b=*/false, b,
            /*c_mod=*/(short)0, acc, /*reuse_a=*/false, /*reuse_b=*/false);
    }

    const int rbase = row0 + ((lane >> 4) << 3);   // +8 row offset for upper lanes
    float* __restrict__ outp = xw + (size_t)rbase * OUT_CH + col0 + mn;
#pragma unroll
    for (int r = 0; r < 8; ++r) outp[(size_t)r * OUT_CH] = acc[r];
}

// ---------------------------------------------------------------------------
// 5) seed accumulator with the self-loop term: out[i] = xw[i] * dinv[i]^2
__global__ void gcn_self_init(const float4* __restrict__ xw4,
                              const float* __restrict__ dinv,
                              float4* __restrict__ out4, int N) {
    int idx = blockIdx.x * blockDim.x + threadIdx.x;   // one float4 per thread
    int node = idx >> 5;                               // 32 float4 per node (128 ch)
    if (node >= N) return;
    float s  = dinv[node];
    float nn = s * s;
    float4 v = xw4[idx];
    v.x *= nn; v.y *= nn; v.z *= nn; v.w *= nn;
    out4[idx] = v;
}

// 6) edge scatter: one wave per edge, lane handles 4 channels.
//    Coalesced b128 gather of xw[src], hardware f32 atomics into out[dst].
__global__ void __launch_bounds__(256)
gcn_scatter(const int* __restrict__ src, const int* __restrict__ dst,
            const float* __restrict__ dinv, const float4* __restrict__ xw4,
            float* __restrict__ out, int E) {
    int gid  = blockIdx.x * blockDim.x + threadIdx.x;
    int e    = gid >> 5;
    int lane = gid & 31;
    if (e >= E) return;
    int s = src[e];
    int d = dst[e];
    __builtin_prefetch((const void*)(xw4 + (size_t)s * 32 + lane), 0, 0);
    float nrm = dinv[s] * dinv[d];
    float4 v = xw4[(size_t)s * 32 + lane];
    float* o = out + (size_t)d * OUT_CH + lane * 4;
    unsafeAtomicAdd(o + 0, v.x * nrm);
    unsafeAtomicAdd(o + 1, v.y * nrm);
    unsafeAtomicAdd(o + 2, v.z * nrm);
    unsafeAtomicAdd(o + 3, v.w * nrm);
}

// 7) out = relu(out + b)
__global__ void gcn_bias_relu(float4* __restrict__ out4, const float4* __restrict__ b4, int N) {
    int idx = blockIdx.x * blockDim.x + threadIdx.x;
    if ((idx >> 5) >= N) return;
    float4 v = out4[idx];
    float4 b = b4[idx & 31];
    v.x = fmaxf(v.x + b.x, 0.0f);
    v.y = fmaxf(v.y + b.y, 0.0f);
    v.z = fmaxf(v.z + b.z, 0.0f);
    v.w = fmaxf(v.w + b.w, 0.0f);
    out4[idx] = v;
}

// ---------------------------------------------------------------------------
extern "C" void kernel_launch(void* const* d_in, const int* in_sizes, int n_in,
                              void* d_out, int out_size, void* d_ws, size_t ws_size,
                              hipStream_t stream) {
    const float* x     = (const float*)d_in[0];
    const int*   edges = (const int*)d_in[1];     // (2, E): row 0 = src, row 1 = dst
    const float* W     = (const float*)d_in[2];
    const float* b     = (const float*)d_in[3];
    float*       out   = (float*)d_out;

    const int N = in_sizes[0] / IN_CH;            // 100000
    const int E = in_sizes[1] / 2;                // 1600000
    const int* src = edges;
    const int* dst = edges + E;

    // workspace layout: xw [N*128 f32] | deg [N u32] | dinv [N f32]
    float*    xw   = (float*)d_ws;
    unsigned* deg  = (unsigned*)(xw + (size_t)N * OUT_CH);
    float*    dinv = (float*)(deg + N);

    const int T = 256;
    gcn_deg_init <<<(N + T - 1) / T, T, 0, stream>>>(deg, N);
    gcn_deg_accum<<<(E + T - 1) / T, T, 0, stream>>>(dst, deg, E);
    gcn_dinv     <<<(N + T - 1) / T, T, 0, stream>>>(deg, dinv, N);

    gcn_xw_wmma  <<<(N + 15) / 16, T, 0, stream>>>(x, W, xw, N);

    const long long nvec = (long long)N * 32;     // float4 elements of out
    gcn_self_init<<<(unsigned)((nvec + T - 1) / T), T, 0, stream>>>(
        (const float4*)xw, dinv, (float4*)out, N);

    const long long esch = (long long)E * 32;     // one wave (32 lanes) per edge
    gcn_scatter  <<<(unsigned)((esch + T - 1) / T), T, 0, stream>>>(
        src, dst, dinv, (const float4*)xw, out, E);

    gcn_bias_relu<<<(unsigned)((nvec + T - 1) / T), T, 0, stream>>>(
        (float4*)out, (const float4*)b, N);
}